// PointPillarScatter_62096637165778
// MI455X (gfx1250) — compile-verified
//
#include <hip/hip_runtime.h>
#include <hip/hip_bf16.h>
#include <stdint.h>

#define BATCH 8
#define NUM_BEV_FEATURES 64
#define NX 256
#define NY 256
#define PLANE (NY * NX)              // 65536 elements per channel plane
#define PILLARS_PER_BLOCK 8          // 8 pillars * 1 wave32 each = 256 threads

// ---------------------------------------------------------------------------
// Kernel 1: zero the canvas, one b128 store per thread (no loop overhead).
// Canvas = 8*64*256*256 floats = 33,554,432 floats = 8,388,608 float4.
// Grid is launched to cover n4 exactly; guard handles any remainder.
// ---------------------------------------------------------------------------
__global__ void zero_canvas_kernel(float4* __restrict__ out, size_t n4) {
    const size_t i = (size_t)blockIdx.x * blockDim.x + threadIdx.x;
    if (i < n4) {
        out[i] = make_float4(0.f, 0.f, 0.f, 0.f);   // global_store_b128
    }
}

// ---------------------------------------------------------------------------
// Kernel 2: scatter pillar rows into the transposed BEV canvas.
//   out[((b*64 + c)*NY + y)*NX + x] = feats[p*64 + c]
// One wave32 per pillar; each lane handles channels (2*lane, 2*lane+1).
// Feature rows are staged global -> LDS with the CDNA5 async-to-LDS path
// (ASYNCcnt), overlapping the coords load + scatter address math with the
// copy. Thread tid stages bytes [tid*8, tid*8+8) and later reads back
// exactly those bytes (floats 2*tid, 2*tid+1 of the block's 512-float
// tile), so a per-wave s_wait_asynccnt suffices — no workgroup barrier.
// ---------------------------------------------------------------------------
__global__ void pillar_scatter_kernel(const float* __restrict__ feats,
                                      const int*   __restrict__ coords,
                                      float*       __restrict__ out,
                                      int P) {
    __shared__ float smem[PILLARS_PER_BLOCK * NUM_BEV_FEATURES]; // 2 KB

    const int tid     = threadIdx.x;            // 0..255
    const int warp    = tid >> 5;               // wave32 id: 0..7
    const int lane    = tid & 31;
    const int pillar0 = blockIdx.x * PILLARS_PER_BLOCK;
    const int p       = pillar0 + warp;

    // ---- async stage: 256 threads x 8 bytes = full 8x64 f32 tile ----------
    // Generic pointer to __shared__ carries the LDS byte offset in its low
    // 32 bits on gfx1250 (aperture-based flat addressing).
    uint32_t lds_addr = (uint32_t)(uintptr_t)(&smem[0]) + (uint32_t)tid * 8u;
    const float* gsrc = feats + (size_t)pillar0 * NUM_BEV_FEATURES + (size_t)tid * 2;

    if (pillar0 + PILLARS_PER_BLOCK <= P) {
        asm volatile("global_load_async_to_lds_b64 %0, %1, off"
                     :: "v"(lds_addr), "v"(gsrc)
                     : "memory");
    } else if (p < P) {
        // ragged tail (not hit for P=160000, kept for safety)
        smem[tid * 2 + 0] = gsrc[0];
        smem[tid * 2 + 1] = gsrc[1];
    }

    // ---- overlap: coords load + scatter address math ----------------------
    // All 32 lanes of a wave load the same 3 ints; the memory system
    // coalesces same-address lanes into one request, so this is cheap and
    // fully overlapped with the async copy above.
    size_t base = 0;
    if (p < P) {
        const int b = coords[3 * p + 0];
        const int y = coords[3 * p + 1];
        const int x = coords[3 * p + 2];
        base = (size_t)b * (NUM_BEV_FEATURES * (size_t)PLANE)
             + (size_t)y * NX + (size_t)x;
    }

    // ---- wait for this wave's async-to-LDS transfers ----------------------
    asm volatile("s_wait_asynccnt 0x0" ::: "memory");

    if (p < P) {
        // ds_load_b64: the two channels this lane owns
        const float2 f = *(const float2*)&smem[warp * NUM_BEV_FEATURES + lane * 2];
        const int c0 = 2 * lane;
        // Two 4-byte scattered stores, stride PLANE between channels.
        // The 134 MB canvas fits in the 192 MB L2, so these coalesce there
        // and HBM sees only the final streaming writeback.
        out[base + (size_t)(c0    ) * PLANE] = f.x;
        out[base + (size_t)(c0 + 1) * PLANE] = f.y;
    }
}

// ---------------------------------------------------------------------------
// Launch
// ---------------------------------------------------------------------------
extern "C" void kernel_launch(void* const* d_in, const int* in_sizes, int n_in,
                              void* d_out, int out_size, void* d_ws, size_t ws_size,
                              hipStream_t stream) {
    const float* feats  = (const float*)d_in[0];   // [P, 64] f32
    const int*   coords = (const int*)d_in[1];     // [P, 3]  i32 (b, y, x)
    float*       out    = (float*)d_out;           // [8, 64, 256, 256] f32

    const int P = in_sizes[1] / 3;                 // 160000

    // 1) zero the canvas (out_size = 33,554,432 floats, divisible by 4)
    {
        const size_t n4 = (size_t)out_size / 4;    // 8,388,608 float4
        const int threads = 256;
        const int blocks  = (int)((n4 + threads - 1) / threads); // 32768
        zero_canvas_kernel<<<blocks, threads, 0, stream>>>((float4*)out, n4);
    }

    // 2) scatter
    {
        const int threads = PILLARS_PER_BLOCK * 32;                          // 256
        const int blocks  = (P + PILLARS_PER_BLOCK - 1) / PILLARS_PER_BLOCK; // 20000
        pillar_scatter_kernel<<<blocks, threads, 0, stream>>>(feats, coords, out, P);
    }
}